// BatchHardTripletLoss_1236950581906
// MI455X (gfx1250) — compile-verified
//
#include <hip/hip_runtime.h>
#include <math.h>

typedef float v2f __attribute__((ext_vector_type(2)));
typedef float v8f __attribute__((ext_vector_type(8)));

#define N 8192
#define D 256
#define MARGIN 0.3f
#define EPS 1e-12f
#define LDSTRIDE 260          // 256 + 4 pad: 8B aligned, spreads LDS banks
#define ITILES 2
#define ROWS_PER_BLOCK (16 * ITILES)

// ---------------- Kernel 1: L2-normalize rows, compute per-row squared norm ----------------
__global__ void norm_kernel(const float* __restrict__ x, float* __restrict__ nf,
                            float* __restrict__ sq) {
    int row  = blockIdx.x * 8 + (threadIdx.x >> 5);   // one wave32 per row
    int lane = threadIdx.x & 31;
    const float* xr = x + (size_t)row * D;
    float v[8];
    float ss = 0.0f;
#pragma unroll
    for (int t = 0; t < 8; ++t) { v[t] = xr[lane * 8 + t]; ss += v[t] * v[t]; }
#pragma unroll
    for (int off = 16; off > 0; off >>= 1) ss += __shfl_xor(ss, off, 32);
    float scale = 1.0f / fmaxf(sqrtf(ss), 1e-12f);
    float* out = nf + (size_t)row * D;
#pragma unroll
    for (int t = 0; t < 8; ++t) out[lane * 8 + t] = v[t] * scale;
    if (lane == 0) sq[row] = ss * scale * scale;
}

// ---------------- Kernel 2: fused Gram (fp32 WMMA) + batch-hard reduction ----------------
__global__ void triplet_kernel(const float* __restrict__ nf,
                               const float* __restrict__ sq,
                               const int* __restrict__ labels,
                               float* __restrict__ loss) {
    __shared__ float As[ROWS_PER_BLOCK * LDSTRIDE];
    __shared__ float sh_hp[8][ROWS_PER_BLOCK];
    __shared__ float sh_hn[8][ROWS_PER_BLOCK];
    __shared__ float sqi_sh[ROWS_PER_BLOCK];
    __shared__ int   lab_sh[ROWS_PER_BLOCK];

    const int i0   = blockIdx.x * ROWS_PER_BLOCK;
    const int tid  = threadIdx.x;
    const int wave = tid >> 5;
    const int lane = tid & 31;
    const int half = lane >> 4;   // 0: K=0,1 / rows 0-7 of C ; 1: K=2,3 / rows 8-15
    const int n    = lane & 15;   // A/B fragment row, C column

    // cooperative load of the 32-row A tile into padded LDS
    for (int idx = tid; idx < ROWS_PER_BLOCK * D; idx += 256) {
        int r = idx >> 8;         // / 256
        int c = idx & 255;
        As[r * LDSTRIDE + c] = nf[(size_t)(i0 + r) * D + c];
    }
    if (tid < ROWS_PER_BLOCK) {
        sqi_sh[tid] = sq[i0 + tid];
        lab_sh[tid] = labels[i0 + tid];
    }
    __syncthreads();

    float hp[ITILES][8], hn[ITILES][8];
#pragma unroll
    for (int t = 0; t < ITILES; ++t)
#pragma unroll
        for (int r = 0; r < 8; ++r) { hp[t][r] = 0.0f; hn[t][r] = INFINITY; }

    // 8 waves interleave over the 512 j-tiles
    for (int jt = wave; jt < N / 16; jt += 8) {
        const int j0 = jt * 16;
        const float* bptr = nf + (size_t)(j0 + n) * D + 2 * half;

        v8f acc[ITILES];
#pragma unroll
        for (int t = 0; t < ITILES; ++t)
#pragma unroll
            for (int r = 0; r < 8; ++r) acc[t][r] = 0.0f;

#pragma unroll 8
        for (int k0 = 0; k0 < D; k0 += 4) {
            v2f b = *(const v2f*)(bptr + k0);             // global_load_b64, L2 resident
#pragma unroll
            for (int t = 0; t < ITILES; ++t) {
                v2f a = *(const v2f*)(&As[(n + 16 * t) * LDSTRIDE + k0 + 2 * half]); // ds_load_b64
                acc[t] = __builtin_amdgcn_wmma_f32_16x16x4_f32(
                    /*neg_a=*/false, a, /*neg_b=*/false, b,
                    /*c_mod=*/(short)0, acc[t], /*reuse_a=*/false, /*reuse_b=*/false);
            }
        }

        const int   lab_j = labels[j0 + n];
        const float sq_j  = sq[j0 + n];
#pragma unroll
        for (int t = 0; t < ITILES; ++t)
#pragma unroll
            for (int r = 0; r < 8; ++r) {
                int   irow = 16 * t + 8 * half + r;       // C/D layout: lane half selects row block
                float d2   = sqi_sh[irow] + sq_j - 2.0f * acc[t][r];
                float d    = sqrtf(fmaxf(d2, EPS));
                bool  pos  = (lab_sh[irow] == lab_j);
                hp[t][r] = fmaxf(hp[t][r], pos ? d : 0.0f);
                hn[t][r] = fminf(hn[t][r], pos ? INFINITY : d);
            }
    }

    // reduce over the 16 columns held by each lane-half (max/min: order independent)
#pragma unroll
    for (int t = 0; t < ITILES; ++t)
#pragma unroll
        for (int r = 0; r < 8; ++r) {
            float a = hp[t][r], b = hn[t][r];
#pragma unroll
            for (int off = 8; off > 0; off >>= 1) {
                a = fmaxf(a, __shfl_xor(a, off, 32));
                b = fminf(b, __shfl_xor(b, off, 32));
            }
            hp[t][r] = a; hn[t][r] = b;
        }

    if (n == 0) {  // lane 0 holds rows 16t+0..7, lane 16 holds rows 16t+8..15
#pragma unroll
        for (int t = 0; t < ITILES; ++t)
#pragma unroll
            for (int r = 0; r < 8; ++r) {
                int irow = 16 * t + 8 * half + r;
                sh_hp[wave][irow] = hp[t][r];
                sh_hn[wave][irow] = hn[t][r];
            }
    }
    __syncthreads();

    if (tid < ROWS_PER_BLOCK) {
        float a = sh_hp[0][tid], b = sh_hn[0][tid];
#pragma unroll
        for (int w = 1; w < 8; ++w) {
            a = fmaxf(a, sh_hp[w][tid]);
            b = fminf(b, sh_hn[w][tid]);
        }
        loss[i0 + tid] = fmaxf(a - b + MARGIN, 0.0f);
    }
}

// ---------------- Kernel 3: deterministic mean over 8192 losses ----------------
__global__ void reduce_kernel(const float* __restrict__ loss, float* __restrict__ out) {
    __shared__ float sh[256];
    int tid = threadIdx.x;
    float s = 0.0f;
    for (int i = tid; i < N; i += 256) s += loss[i];   // fixed per-thread order
    sh[tid] = s;
    __syncthreads();
    for (int off = 128; off > 0; off >>= 1) {
        if (tid < off) sh[tid] += sh[tid + off];
        __syncthreads();
    }
    if (tid == 0) out[0] = sh[0] / (float)N;
}

extern "C" void kernel_launch(void* const* d_in, const int* in_sizes, int n_in,
                              void* d_out, int out_size, void* d_ws, size_t ws_size,
                              hipStream_t stream) {
    const float* features = (const float*)d_in[0];
    const int*   labels   = (const int*)d_in[1];

    float* nf   = (float*)d_ws;          // 8192*256 floats (8 MB)
    float* sq   = nf + (size_t)N * D;    // 8192 floats
    float* lossb= sq + N;                // 8192 floats

    norm_kernel<<<N / 8, 256, 0, stream>>>(features, nf, sq);
    triplet_kernel<<<N / ROWS_PER_BLOCK, 256, 0, stream>>>(nf, sq, labels, lossb);
    reduce_kernel<<<1, 256, 0, stream>>>(lossb, (float*)d_out);
}